// knn_mask_interp_6201932775996
// MI455X (gfx1250) — compile-verified
//
#include <hip/hip_runtime.h>
#include <math.h>

typedef __attribute__((ext_vector_type(2))) float v2f;
typedef __attribute__((ext_vector_type(8))) float v8f;

#define M_PTS 8192
#define N_PTS 32768
#define D_MASK 32
#define K_NN 8

// ---------------------------------------------------------------------------
// Prep: src4[m] = (-2*sx, -2*sy, -2*sz, ||s||^2)
// One V_WMMA_F32_16X16X4_F32 then computes d'(n,m) = ||s||^2 - 2 q.s for a
// 16x16 tile (||q||^2 dropped: constant per query, argmin-order invariant).
// ---------------------------------------------------------------------------
__global__ void knn_prep_kernel(const float* __restrict__ s_coor,
                                float4* __restrict__ src4) {
  int m = blockIdx.x * blockDim.x + threadIdx.x;
  if (m < M_PTS) {
    float x = s_coor[m * 3 + 0];
    float y = s_coor[m * 3 + 1];
    float z = s_coor[m * 3 + 2];
    src4[m] = make_float4(-2.0f * x, -2.0f * y, -2.0f * z, x * x + y * y + z * z);
  }
}

__device__ __forceinline__ void topk_insert(float cand, int cidx,
                                            float (&dist)[K_NN], int (&didx)[K_NN]) {
  // dist[] sorted descending: dist[0] is the worst of the kept top-8.
  if (cand < dist[0]) {
    dist[0] = cand;
    didx[0] = cidx;
#pragma unroll
    for (int j = 0; j < K_NN - 1; ++j) {
      if (dist[j] < dist[j + 1]) {
        float td = dist[j]; dist[j] = dist[j + 1]; dist[j + 1] = td;
        int   ti = didx[j]; didx[j] = didx[j + 1]; didx[j + 1] = ti;
      }
    }
  }
}

// ---------------------------------------------------------------------------
// Main: 256 threads = 8 wave32; each wave owns 16 queries.
// Whole preprocessed source set (128 KB) staged once in LDS (320 KB/WGP)
// via the CDNA5 async global->LDS path (ASYNCcnt). Inner loop: 2 pipelined
// WMMAs per iteration -> 512 distances, top-8 maintenance gated by a single
// min-tree compare + branch per 32 sources.
// ---------------------------------------------------------------------------
__global__ __launch_bounds__(256) void knn_main_kernel(
    const float4* __restrict__ src4,
    const float*  __restrict__ s_mask,
    const float*  __restrict__ d_coor,
    float*        __restrict__ out) {
  extern __shared__ float4 smem[];  // M_PTS float4 = 128 KB
  const int tid = threadIdx.x;

  // Cooperative stage of all sources into LDS.
#if defined(__gfx1250__)
  for (int i = tid; i < M_PTS; i += 256) {
    unsigned long long ga = (unsigned long long)(uintptr_t)(src4 + i);
    // Low 32 bits of a generic LDS pointer = workgroup-relative LDS byte addr.
    unsigned int la = (unsigned int)(uintptr_t)(smem + i);
    asm volatile("global_load_async_to_lds_b128 %0, %1, off"
                 :: "v"(la), "v"(ga)
                 : "memory");
  }
  asm volatile("s_wait_asynccnt 0" ::: "memory");
#else
  for (int i = tid; i < M_PTS; i += 256) smem[i] = src4[i];
#endif
  __syncthreads();

  const int  lane  = tid & 31;
  const int  wave  = tid >> 5;
  const int  qbase = (blockIdx.x * 8 + wave) * 16;
  const int  q     = lane & 15;
  const bool hi    = lane >= 16;

  // B tile (4x16 queries): lanes 0-15 -> K0(qx),K1(qy); lanes 16-31 -> K2(qz),K3(1)
  const float qx = d_coor[(qbase + q) * 3 + 0];
  const float qy = d_coor[(qbase + q) * 3 + 1];
  const float qz = d_coor[(qbase + q) * 3 + 2];
  v2f b;
  b[0] = hi ? qz : qx;
  b[1] = hi ? 1.0f : qy;

  float dist[K_NN];
  int   didx[K_NN];
#pragma unroll
  for (int j = 0; j < K_NN; ++j) { dist[j] = 3.4e38f; didx[j] = 0; }

  const v8f czero = {};

  // Software-pipelined A-tile fetch from LDS, 2 chunks (32 sources) per
  // iteration: two independent WMMAs (same B, disjoint A/D) pipeline
  // back-to-back; one min-tree + one branch gates all 16 candidates.
  float4 s0 = smem[q];
  float4 s1 = smem[16 + q];
  for (int m = 0; m < M_PTS; m += 32) {
    float4 p0 = smem[(((m + 32) & (M_PTS - 1))) + q];
    float4 p1 = smem[(((m + 48) & (M_PTS - 1))) + q];

    // A tiles (16 sources x 4): lanes 0-15 -> K0=-2sx,K1=-2sy;
    //                           lanes 16-31 -> K2=-2sz,K3=||s||^2
    v2f a0, a1;
    a0[0] = hi ? s0.z : s0.x;
    a0[1] = hi ? s0.w : s0.y;
    a1[0] = hi ? s1.z : s1.x;
    a1[1] = hi ? s1.w : s1.y;

    v8f d0 = __builtin_amdgcn_wmma_f32_16x16x4_f32(
        false, a0, false, b, (short)0, czero, false, false);
    v8f d1 = __builtin_amdgcn_wmma_f32_16x16x4_f32(
        false, a1, false, b, (short)0, czero, false, false);

    // Single gate for all 16 candidates: min-tree vs current 8th-best.
    float ma = fminf(fminf(fminf(d0[0], d0[1]), fminf(d0[2], d0[3])),
                     fminf(fminf(d0[4], d0[5]), fminf(d0[6], d0[7])));
    float mb = fminf(fminf(fminf(d1[0], d1[1]), fminf(d1[2], d1[3])),
                     fminf(fminf(d1[4], d1[5]), fminf(d1[6], d1[7])));
    float cmin = fminf(ma, mb);

    if (cmin < dist[0]) {
      const int mb0 = m + (hi ? 8 : 0);
      const int mb1 = m + 16 + (hi ? 8 : 0);
#pragma unroll
      for (int v = 0; v < 8; ++v) {
        topk_insert(d0[v], mb0 + v, dist, didx);
      }
#pragma unroll
      for (int v = 0; v < 8; ++v) {
        topk_insert(d1[v], mb1 + v, dist, didx);
      }
    }
    s0 = p0;
    s1 = p1;
  }

  // Merge the two half-lane top-8 lists for this query (disjoint source sets).
#pragma unroll
  for (int j = 0; j < K_NN; ++j) {
    float pd = __shfl_xor(dist[j], 16, 32);
    int   pi = __shfl_xor(didx[j], 16, 32);
    topk_insert(pd, pi, dist, didx);
  }

  // Epilogue: binary-mask mode over K = (ones >= 5).
  // lane q -> mask cols 0..15 ; lane q+16 -> cols 16..31.
  const int coff = hi ? 16 : 0;
  float4 acc0 = make_float4(0.f, 0.f, 0.f, 0.f);
  float4 acc1 = acc0, acc2 = acc0, acc3 = acc0;
#pragma unroll
  for (int k = 0; k < K_NN; ++k) {
    const float4* row = (const float4*)(s_mask + (size_t)didx[k] * D_MASK + coff);
    float4 r0 = row[0], r1 = row[1], r2 = row[2], r3 = row[3];
    acc0.x += r0.x; acc0.y += r0.y; acc0.z += r0.z; acc0.w += r0.w;
    acc1.x += r1.x; acc1.y += r1.y; acc1.z += r1.z; acc1.w += r1.w;
    acc2.x += r2.x; acc2.y += r2.y; acc2.z += r2.z; acc2.w += r2.w;
    acc3.x += r3.x; acc3.y += r3.y; acc3.z += r3.z; acc3.w += r3.w;
  }
  float4* po = (float4*)(out + (size_t)(qbase + q) * D_MASK + coff);
  po[0] = make_float4(acc0.x > 4.f ? 1.f : 0.f, acc0.y > 4.f ? 1.f : 0.f,
                      acc0.z > 4.f ? 1.f : 0.f, acc0.w > 4.f ? 1.f : 0.f);
  po[1] = make_float4(acc1.x > 4.f ? 1.f : 0.f, acc1.y > 4.f ? 1.f : 0.f,
                      acc1.z > 4.f ? 1.f : 0.f, acc1.w > 4.f ? 1.f : 0.f);
  po[2] = make_float4(acc2.x > 4.f ? 1.f : 0.f, acc2.y > 4.f ? 1.f : 0.f,
                      acc2.z > 4.f ? 1.f : 0.f, acc2.w > 4.f ? 1.f : 0.f);
  po[3] = make_float4(acc3.x > 4.f ? 1.f : 0.f, acc3.y > 4.f ? 1.f : 0.f,
                      acc3.z > 4.f ? 1.f : 0.f, acc3.w > 4.f ? 1.f : 0.f);
}

extern "C" void kernel_launch(void* const* d_in, const int* in_sizes, int n_in,
                              void* d_out, int out_size, void* d_ws, size_t ws_size,
                              hipStream_t stream) {
  const float* s_coor = (const float*)d_in[0];  // [M,3]
  const float* s_mask = (const float*)d_in[1];  // [M,32]
  const float* d_coor = (const float*)d_in[2];  // [N,3]
  float* out = (float*)d_out;                   // [N,32]
  float4* src4 = (float4*)d_ws;                 // 128 KB scratch

  knn_prep_kernel<<<M_PTS / 256, 256, 0, stream>>>(s_coor, src4);
  knn_main_kernel<<<N_PTS / (16 * 8), 256, M_PTS * sizeof(float4), stream>>>(
      src4, s_mask, d_coor, out);
}